// GridSampleDAS_45603962749239
// MI455X (gfx1250) — compile-verified
//
#include <hip/hip_runtime.h>
#include <cstdint>
#include <cstddef>

// ---------------------------------------------------------------------------
// DAS beamforming (grid_sample-style fp16-quantized delays) for MI455X gfx1250.
//
// Strategy:
//   * 1 block = (a, 4 z-rows). 256 threads (8 wave32). Thread t owns pixel
//     column x=t for 4 consecutive z rows -> 1024 pixels per block.
//   * RF traces rf[a,e,:] (8KB each) are staged into LDS with the CDNA5
//     async global->LDS DMA path (global_load_async_to_lds_b128, ASYNCcnt),
//     4 traces (32KB) per stage, double buffered (64KB LDS total),
//     overlapping next-stage DMA with current-stage compute.
//   * Interpolation gathers come from LDS (ds_load_b32).
// ---------------------------------------------------------------------------

namespace {
constexpr int   A_   = 5;
constexpr int   E_   = 128;
constexpr int   S_   = 2048;
constexpr int   NZ_  = 400;
constexpr int   NX_  = 256;
constexpr float C0_  = 1540.0f;
constexpr float FS_  = 20.0e6f;   // DEC == 1

constexpr int THREADS_ = 256;     // 8 wave32
constexpr int ZT_      = 4;       // z-rows per block
constexpr int CH_      = 4;       // traces per pipeline stage
constexpr int NSTAGE_  = E_ / CH_;                 // 32
constexpr int STAGE_FLOATS_ = CH_ * S_;            // 8192 floats = 32KB
constexpr int BYTES_PER_THREAD_ = STAGE_FLOATS_ * 4 / THREADS_;  // 128
constexpr int B128_PER_THREAD_  = BYTES_PER_THREAD_ / 16;        // 8
} // namespace

// Low 32 bits of a generic (flat) pointer into LDS == LDS byte address
// (flat LDS addresses are {shared-aperture, offset[31:0]} per CDNA5 ISA).
__device__ __forceinline__ unsigned lds_byte_addr(const void* p) {
    return (unsigned)(unsigned long long)(uintptr_t)p;
}

// Issue this thread's slice (128B as 8x b128) of a 32KB stage copy,
// global -> LDS, via the CDNA5 async DMA path (tracked by ASYNCcnt).
__device__ __forceinline__ void async_stage_copy(unsigned ldsBase,
                                                 const float* __restrict__ gsrc,
                                                 unsigned tid) {
    unsigned long long gbase = (unsigned long long)gsrc;
#pragma unroll
    for (int j = 0; j < B128_PER_THREAD_; ++j) {
        unsigned off = tid * (unsigned)BYTES_PER_THREAD_ + (unsigned)(j * 16);
        unsigned lds = ldsBase + off;
        // GVS mode: mem_addr = SGPR64 + VGPR32 offset ; vdst holds LDS address
        asm volatile("global_load_async_to_lds_b128 %0, %1, %2"
                     :: "v"(lds), "v"(off), "s"(gbase)
                     : "memory");
    }
}

__device__ __forceinline__ void wait_asynccnt0() {
#if __has_builtin(__builtin_amdgcn_s_wait_asynccnt)
    __builtin_amdgcn_s_wait_asynccnt(0);
#else
    asm volatile("s_wait_asynccnt 0x0" ::: "memory");
#endif
}

__global__ __launch_bounds__(THREADS_)
void das_gridsample_kernel(const float* __restrict__ rf,     // [A,E,S]
                           const float* __restrict__ d_tx,   // [A,NZ,NX]
                           const float* __restrict__ d_rx,   // [E,NZ,NX]
                           const float* __restrict__ t0,     // [A]
                           float* __restrict__ out) {        // [A,NZ,NX]
    __shared__ __align__(16) float buf[2][STAGE_FLOATS_];    // 64KB double buffer

    const int tid = (int)threadIdx.x;
    const int bid = (int)blockIdx.x;
    constexpr int ZBLKS = NZ_ / ZT_;          // 100
    const int a  = bid / ZBLKS;
    const int z0 = (bid - a * ZBLKS) * ZT_;
    const int x  = tid;

    const float* rfA = rf + (size_t)a * E_ * S_;
    const float  t0a = t0[a];

    float dtx[ZT_];
    float acc[ZT_];
#pragma unroll
    for (int zz = 0; zz < ZT_; ++zz) {
        dtx[zz] = d_tx[((size_t)a * NZ_ + (z0 + zz)) * NX_ + x];
        acc[zz] = 0.0f;
    }

    const unsigned ldsb[2] = { lds_byte_addr(&buf[0][0]), lds_byte_addr(&buf[1][0]) };

    // Prologue: start DMA of stage 0 (traces e = 0..3).
    async_stage_copy(ldsb[0], rfA, (unsigned)tid);

    constexpr float NORM = 2.0f / (float)(S_ - 1);
    constexpr float HALF_SPAN = 0.5f * (float)(S_ - 1);

    for (int k = 0; k < NSTAGE_; ++k) {
        const int p = k & 1;

        wait_asynccnt0();        // this wave's stage-k DMA landed in LDS
        __syncthreads();         // all waves' DMA landed; all stage-(k-1) reads done

        if (k + 1 < NSTAGE_) {   // overlap: DMA stage k+1 while computing stage k
            async_stage_copy(ldsb[p ^ 1],
                             rfA + (size_t)(k + 1) * CH_ * S_,
                             (unsigned)tid);
            // warm L2/L0 for next stage's d_rx rows
            __builtin_prefetch(&d_rx[((size_t)(k + 1) * CH_ * NZ_ + z0) * NX_ + x], 0, 0);
        }

#pragma unroll
        for (int c = 0; c < CH_; ++c) {
            const int e = k * CH_ + c;
            const float* tr = &buf[p][c * S_];
#pragma unroll
            for (int zz = 0; zz < ZT_; ++zz) {
                const float drx = d_rx[((size_t)e * NZ_ + (z0 + zz)) * NX_ + x];
                {
#pragma clang fp contract(off)
                    // delay in samples (IEEE f32 div by C0, matching reference)
                    float delay = ((dtx[zz] + drx) / C0_ - t0a) * FS_;
                    float gx    = delay * NORM - 1.0f;
                    // replicate the fp16 round-trip quantization (RNE)
                    _Float16 gh = (_Float16)gx;
                    float gq    = (float)gh;
                    // align_corners=True pixel coordinate
                    float ix  = (gq + 1.0f) * 0.5f * (float)(S_ - 1);
                    float x0f = floorf(ix);
                    float w   = ix - x0f;
                    int xi0 = (int)x0f;
                    int xi1 = xi0 + 1;
                    int c0 = xi0 < 0 ? 0 : (xi0 > S_ - 1 ? S_ - 1 : xi0);
                    int c1 = xi1 < 0 ? 0 : (xi1 > S_ - 1 ? S_ - 1 : xi1);
                    float s0 = tr[c0];                 // LDS gather
                    float s1 = tr[c1];                 // LDS gather
                    s0 = (xi0 >= 0 && xi0 < S_) ? s0 : 0.0f;  // zero padding
                    s1 = (xi1 >= 0 && xi1 < S_) ? s1 : 0.0f;
                    acc[zz] = acc[zz] + ((1.0f - w) * s0 + w * s1);
                }
            }
        }
        (void)HALF_SPAN;
    }

#pragma unroll
    for (int zz = 0; zz < ZT_; ++zz) {
        out[((size_t)a * NZ_ + (z0 + zz)) * NX_ + x] = acc[zz];
    }
}

extern "C" void kernel_launch(void* const* d_in, const int* in_sizes, int n_in,
                              void* d_out, int out_size, void* d_ws, size_t ws_size,
                              hipStream_t stream) {
    (void)in_sizes; (void)n_in; (void)d_ws; (void)ws_size; (void)out_size;
    const float* rf   = (const float*)d_in[0];   // [A,E,S] f32
    const float* dtx  = (const float*)d_in[1];   // [A,NZ,NX] f32
    const float* drx  = (const float*)d_in[2];   // [E,NZ,NX] f32
    const float* t0v  = (const float*)d_in[3];   // [A] f32
    float* out = (float*)d_out;                  // [A,NZ,NX] f32

    dim3 grid(A_ * (NZ_ / ZT_));                 // 500 blocks
    dim3 block(THREADS_);                        // 8 wave32
    das_gridsample_kernel<<<grid, block, 0, stream>>>(rf, dtx, drx, t0v, out);
}